// NetGraphSage_85358180040740
// MI455X (gfx1250) — compile-verified
//
#include <hip/hip_runtime.h>
#include <hip/hip_bf16.h>
#include <math.h>

// GraphSAGE (2x SAGEConv + global_mean_pool + FC + sigmoid) for MI455X/gfx1250.
// fp32 end-to-end; GEMMs use V_WMMA_F32_16X16X4_F32 (exact fp32 on the matrix pipe).
// edge_index / batch are declared int64 in the reference -> cast to long long.

typedef __attribute__((ext_vector_type(2))) float v2f;
typedef __attribute__((ext_vector_type(8))) float v8f;

#define FEAT 128
#define NGRAPHS 256
#define LDSA_STRIDE 260   // 16*? pad: 260 % 64 banks = 4 -> conflict-free column reads

// ---------------- scratch zeroing ----------------
__global__ __launch_bounds__(256) void zero_f4(float* __restrict__ p, int n) {
    int i = (blockIdx.x * 256 + threadIdx.x) * 4;
    if (i < n) {
        float4 z = {0.f, 0.f, 0.f, 0.f};
        *(float4*)(p + i) = z;
    }
}

// ---------------- degree / graph counts ----------------
__global__ __launch_bounds__(256) void edge_count(const long long* __restrict__ ei,
                                                  int nE, float* __restrict__ cnt) {
    int e = blockIdx.x * 256 + threadIdx.x;
    if (e < nE) {
        int d = (int)ei[nE + e];            // dst row
        atomicAdd(cnt + d, 1.0f);
    }
}

__global__ __launch_bounds__(256) void graph_count(const long long* __restrict__ batch,
                                                   int nN, float* __restrict__ gcnt) {
    int i = blockIdx.x * 256 + threadIdx.x;
    if (i < nN) atomicAdd(gcnt + (int)batch[i], 1.0f);
}

// ---------------- edge scatter: aggr[dst] += feat[src] ----------------
// one wave32 per edge; lane handles 4 contiguous features (float4 gather, f32 atomics).
__global__ __launch_bounds__(256) void scatter_edges(const float* __restrict__ feat,
                                                     const long long* __restrict__ ei,
                                                     float* __restrict__ aggr, int nE) {
    int e = blockIdx.x * 8 + (threadIdx.x >> 5);
    if (e >= nE) return;
    int lane = threadIdx.x & 31;
    int s = (int)ei[e];
    int d = (int)ei[nE + e];
    const float4 v = *(const float4*)(feat + (size_t)s * FEAT + lane * 4);
    float* dst = aggr + (size_t)d * FEAT + lane * 4;
    atomicAdd(dst + 0, v.x);
    atomicAdd(dst + 1, v.y);
    atomicAdd(dst + 2, v.z);
    atomicAdd(dst + 3, v.w);
}

// ---------------- fused SAGE GEMM: OUT = act(concat[X, AGG/max(cnt,1)] @ W) ------
// Block: 256 thr = 8 waves. 16-row tile (K=256) staged in LDS; wave w owns output
// cols [16w,16w+16). K stepped by 4 with V_WMMA_F32_16X16X4_F32.
// Optional fused mean-pool accumulation (layer 2): gsum[batch[row]] += out.
__global__ __launch_bounds__(256) void sage_gemm_wmma(
    const float* __restrict__ X,      // [N,128] self features
    const float* __restrict__ AGG,    // [N,128] neighbor sums
    const float* __restrict__ CNT,    // [N] in-degree
    const float* __restrict__ W,      // [256,128] row-major
    float* __restrict__ OUT,          // [N,128]
    int nN, int do_relu,
    const long long* __restrict__ batch,  // null => no pooling
    float* __restrict__ gsum)             // [256,128]
{
    __shared__ float lds_a[16 * LDSA_STRIDE];

    const int row0 = blockIdx.x * 16;
    const int t = threadIdx.x;

    // ---- stage A tile: thread t loads row (t>>4), 16-col chunk ((t&15)*16) ----
    {
        int r  = t >> 4;
        int c0 = (t & 15) * 16;
        int row = row0 + r; if (row >= nN) row = nN - 1;
        float inv = 1.0f / fmaxf(CNT[row], 1.0f);
        const float* src;
        float scale;
        if (c0 < FEAT) { src = X   + (size_t)row * FEAT + c0;        scale = 1.0f; }
        else           { src = AGG + (size_t)row * FEAT + (c0-FEAT); scale = inv;  }
        float* dstl = &lds_a[r * LDSA_STRIDE + c0];
#pragma unroll
        for (int i = 0; i < 16; i += 4) {
            float4 v = *(const float4*)(src + i);
            dstl[i+0] = v.x * scale;
            dstl[i+1] = v.y * scale;
            dstl[i+2] = v.z * scale;
            dstl[i+3] = v.w * scale;
        }
    }
    __syncthreads();

    const int wave  = t >> 5;
    const int lane  = t & 31;
    const int l15   = lane & 15;
    const int half2 = (lane >> 4) * 2;   // K sub-offset: lanes 0-15 -> {0,1}, 16-31 -> {2,3}
    const int n0    = wave * 16;         // output column tile

    v8f acc = {};
    const float* ap = &lds_a[l15 * LDSA_STRIDE + half2];
    const float* bp = W + (size_t)half2 * FEAT + n0 + l15;

#pragma unroll 8
    for (int k = 0; k < 2 * FEAT; k += 4) {
        v2f a, b;
        a.x = ap[k];
        a.y = ap[k + 1];
        b.x = bp[(size_t)k * FEAT];          // W[k+half2][n0+l15]
        b.y = bp[(size_t)(k + 1) * FEAT];    // W[k+half2+1][n0+l15]
        acc = __builtin_amdgcn_wmma_f32_16x16x4_f32(
            /*neg_a=*/false, a, /*neg_b=*/false, b,
            /*c_mod=*/(short)0, acc, /*reuse_a=*/false, /*reuse_b=*/false);
    }

    // ---- epilogue: C layout VGPR r -> M=r (lanes 0-15), M=r+8 (lanes 16-31) ----
    const int mbase = row0 + (lane >> 4) * 8;
    const int ocol  = n0 + l15;
#pragma unroll
    for (int r = 0; r < 8; ++r) {
        int orow = mbase + r;
        if (orow >= nN) continue;
        float v = acc[r];
        if (do_relu) v = fmaxf(v, 0.0f);
        OUT[(size_t)orow * FEAT + ocol] = v;
        if (gsum) {
            int g = (int)batch[orow];
            atomicAdd(&gsum[(size_t)g * FEAT + ocol], v);
        }
    }
}

// ---------------- pooled mean -> FC -> sigmoid ----------------
__global__ __launch_bounds__(128) void pool_fc_sigmoid(const float* __restrict__ gsum,
                                                       const float* __restrict__ gcnt,
                                                       const float* __restrict__ Wfc,
                                                       float* __restrict__ out) {
    int g = blockIdx.x;
    int t = threadIdx.x;                      // 0..127
    float inv = 1.0f / fmaxf(gcnt[g], 1.0f);
    float v = gsum[(size_t)g * FEAT + t] * inv * Wfc[t];
#pragma unroll
    for (int o = 16; o > 0; o >>= 1) v += __shfl_down(v, o, 32);
    __shared__ float red[4];
    if ((t & 31) == 0) red[t >> 5] = v;
    __syncthreads();
    if (t == 0) {
        float s = red[0] + red[1] + red[2] + red[3];
        out[g] = 1.0f / (1.0f + __expf(-s));
    }
}

extern "C" void kernel_launch(void* const* d_in, const int* in_sizes, int n_in,
                              void* d_out, int out_size, void* d_ws, size_t ws_size,
                              hipStream_t stream) {
    const float*     x     = (const float*)d_in[0];
    const long long* ei    = (const long long*)d_in[1];  // [2,E] int64
    const long long* batch = (const long long*)d_in[2];  // [N] int64
    const float*     W1    = (const float*)d_in[3];      // [256,128]
    const float*     W2    = (const float*)d_in[4];      // [256,128]
    const float*     Wfc   = (const float*)d_in[5];      // [128]
    float* out = (float*)d_out;

    const int E = in_sizes[1] / 2;
    const int N = in_sizes[2];
    const int G = NGRAPHS;

    // scratch layout (floats): regions needing zeroing first (contiguous), h1 last
    float* ws    = (float*)d_ws;
    float* aggr1 = ws;                              // N*128
    float* aggr2 = aggr1 + (size_t)N * FEAT;        // N*128
    float* cnt   = aggr2 + (size_t)N * FEAT;        // N
    float* gsum  = cnt   + N;                       // G*128
    float* gcnt  = gsum  + (size_t)G * FEAT;        // G
    float* h1    = gcnt  + G;                       // N*128 (fully overwritten)

    const int zeroN = 2 * N * FEAT + N + G * FEAT + G;   // multiple of 4
    zero_f4<<<(zeroN / 4 + 255) / 256, 256, 0, stream>>>(ws, zeroN);

    edge_count <<<(E + 255) / 256, 256, 0, stream>>>(ei, E, cnt);
    graph_count<<<(N + 255) / 256, 256, 0, stream>>>(batch, N, gcnt);

    const int tiles = (N + 15) / 16;

    // Layer 1: aggr1 = segsum(x[src] -> dst); h1 = relu(concat[x, aggr1/cnt] @ W1)
    scatter_edges<<<(E + 7) / 8, 256, 0, stream>>>(x, ei, aggr1, E);
    sage_gemm_wmma<<<tiles, 256, 0, stream>>>(x, aggr1, cnt, W1, h1, N, 1, nullptr, nullptr);

    // Layer 2: aggr2 = segsum(h1[src] -> dst); h2 = concat[h1, aggr2/cnt] @ W2
    // (written in place over h1; fused graph-mean accumulation into gsum)
    scatter_edges<<<(E + 7) / 8, 256, 0, stream>>>(h1, ei, aggr2, E);
    sage_gemm_wmma<<<tiles, 256, 0, stream>>>(h1, aggr2, cnt, W2, h1, N, 0, batch, gsum);

    // out[g] = sigmoid( (gsum[g]/gcnt[g]) . Wfc )
    pool_fc_sigmoid<<<G, 128, 0, stream>>>(gsum, gcnt, Wfc, out);
}